// HungarianMatcherLite_68899865362671
// MI455X (gfx1250) — compile-verified
//
#include <hip/hip_runtime.h>

#define BB 128
#define QQ 500
#define QPAD 512          // padded rows so WMMA tile stores are branch-free
#define CC 80
#define TT 64
#define QT (QQ * TT)      // 32000 live elements
#define NTHREADS 256
#define NWAVES (NTHREADS / 32)

typedef __attribute__((ext_vector_type(2))) float v2f;
typedef __attribute__((ext_vector_type(8))) float v8f;

// Dynamic LDS layout (float units)
#define OFF_C      0
#define OFF_REDV   (OFF_C + QPAD * TT)      // 8 wave-partial values
#define OFF_REDI   (OFF_REDV + NWAVES)      // 8 wave-partial indices (int)
#define OFF_IDS    (OFF_REDI + NWAVES)      // 64 target class ids (int)
#define OFF_TRAW   (OFF_IDS + TT)           // 64*4 raw target boxes
#define OFF_TN     (OFF_TRAW + TT * 4)      // 64*4 normalized target boxes
#define OFF_BRAW   (OFF_TN + TT * 4)        // 500*4 raw pred boxes
#define OFF_BN     (OFF_BRAW + QQ * 4)      // 500*4 normalized pred boxes
#define OFF_AREAA  (OFF_BN + QQ * 4)        // 500 pred areas
#define OFF_AREAB  (OFF_AREAA + QQ)         // 64 target areas
#define SMEM_FLOATS (OFF_AREAB + TT)        // 37924 floats = 151696 B < 320 KB

__device__ __forceinline__ float focal_diff(float x) {
    const float ALPHA = 0.25f, EPS = 1e-8f;
    float p   = 1.0f / (1.0f + expf(-x));
    float neg = (1.0f - ALPHA) * p * p * (-log1pf(EPS - p));
    float omp = 1.0f - p;
    float pos = ALPHA * omp * omp * (-logf(p + EPS));
    return pos - neg;
}

__global__ __launch_bounds__(NTHREADS)
void matcher_fused(const float* __restrict__ logits,   // [B,Q,C]
                   const float* __restrict__ boxes,    // [B,Q,4]
                   const int*   __restrict__ ids,      // [B,T]
                   const float* __restrict__ tbox,     // [B,T,4]
                   const float* __restrict__ img,      // [B,4]
                   float* __restrict__ outC,           // [B,Q,T]
                   float* __restrict__ outRows,        // [B,T] (as float)
                   float* __restrict__ outCols)        // [B,T] (as float)
{
    extern __shared__ float smem[];
    float* sC     = smem + OFF_C;
    float* sRedV  = smem + OFF_REDV;
    int*   sRedI  = (int*)(smem + OFF_REDI);
    int*   sIds   = (int*)(smem + OFF_IDS);
    float* sTraw  = smem + OFF_TRAW;
    float* sTn    = smem + OFF_TN;
    float* sBraw  = smem + OFF_BRAW;
    float* sBn    = smem + OFF_BN;
    float* sAreaA = smem + OFF_AREAA;
    float* sAreaB = smem + OFF_AREAB;

    const int b   = blockIdx.x;
    const int tid = threadIdx.x;

    const float s0 = img[b * 4 + 0], s1 = img[b * 4 + 1];
    const float s2 = img[b * 4 + 2], s3 = img[b * 4 + 3];

    // ---- Phase 0: stage targets and pred boxes into LDS; normalize ONCE (== reference) ----
    if (tid < TT) {
        sIds[tid] = ids[b * TT + tid];
        float4 tb = *(const float4*)(tbox + ((size_t)b * TT + tid) * 4);
        sTraw[tid * 4 + 0] = tb.x; sTraw[tid * 4 + 1] = tb.y;
        sTraw[tid * 4 + 2] = tb.z; sTraw[tid * 4 + 3] = tb.w;
        sTn[tid * 4 + 0] = tb.x / s0; sTn[tid * 4 + 1] = tb.y / s1;
        sTn[tid * 4 + 2] = tb.z / s2; sTn[tid * 4 + 3] = tb.w / s3;
        sAreaB[tid] = (tb.z - tb.x) * (tb.w - tb.y);
    }
    for (int q = tid; q < QQ; q += NTHREADS) {
        float4 pb = *(const float4*)(boxes + ((size_t)b * QQ + q) * 4);
        sBraw[q * 4 + 0] = pb.x; sBraw[q * 4 + 1] = pb.y;
        sBraw[q * 4 + 2] = pb.z; sBraw[q * 4 + 3] = pb.w;
        sBn[q * 4 + 0] = pb.x / s0; sBn[q * 4 + 1] = pb.y / s1;
        sBn[q * 4 + 2] = pb.z / s2; sBn[q * 4 + 3] = pb.w / s3;
        sAreaA[q] = (pb.z - pb.x) * (pb.w - pb.y);
    }
    __syncthreads();

    // ---- Phase 1: class cost as exact one-hot GEMM via v_wmma_f32_16x16x4_f32 ----
    // cost_class[q,t] = diff[q, ids[t]] = sum_c diff[q,c] * onehot[c,t]
    // Branch-free: out-of-range rows (500..511) use a clamped row pointer; their
    // (finite) garbage D-rows land in the LDS pad and are never read. A-row m only
    // feeds D-row m, so live rows are exact. Uniform control flow => EXEC all-ones.
    {
        const int wave = tid >> 5, lane = tid & 31;
        const int m = lane & 15, khalf = lane >> 4;   // A/B lane-half K split per ISA 7.12.2
        for (int qt = wave; qt < QPAD / 16; qt += NWAVES) {
            int q  = qt * 16 + m;
            int qc = (q < QQ) ? q : (QQ - 1);
            const float* lrow = logits + ((size_t)b * QQ + qc) * CC;
            v8f acc[4];
            #pragma unroll
            for (int nt = 0; nt < 4; ++nt)
                #pragma unroll
                for (int r = 0; r < 8; ++r) acc[nt][r] = 0.0f;

            for (int k0 = 0; k0 < CC; k0 += 4) {      // 20 chained K-steps
                int ka = k0 + 2 * khalf;              // lanes 0-15: K=k0,k0+1 ; 16-31: k0+2,k0+3
                float2 lv = *(const float2*)(lrow + ka);   // single global_load_b64
                v2f a;
                a.x = focal_diff(lv.x);
                a.y = focal_diff(lv.y);
                #pragma unroll
                for (int nt = 0; nt < 4; ++nt) {      // T=64 -> 4 N-tiles, A reused
                    int cls = sIds[nt * 16 + m];
                    v2f bv;
                    bv.x = (cls == ka)     ? 1.0f : 0.0f;
                    bv.y = (cls == ka + 1) ? 1.0f : 0.0f;
                    acc[nt] = __builtin_amdgcn_wmma_f32_16x16x4_f32(
                        false, a, false, bv, (short)0, acc[nt], false, false);
                }
            }
            // D layout: VGPR r -> M=r (lanes 0-15) / M=r+8 (lanes 16-31), N = lane%16.
            #pragma unroll
            for (int r = 0; r < 8; ++r) {
                int qq2 = qt * 16 + r + 8 * khalf;    // < QPAD always
                #pragma unroll
                for (int nt = 0; nt < 4; ++nt)
                    sC[qq2 * TT + nt * 16 + m] = 2.0f * acc[nt][r];  // CLASS_W = 2
            }
        }
    }
    __syncthreads();

    // ---- Phase 1b: add L1 (pre-normalized) + GIoU (raw); b128 LDS/global traffic ----
    for (int e4 = tid * 4; e4 < QT; e4 += NTHREADS * 4) {
        int q  = e4 >> 6;          // 4 consecutive t share one q (64 % 4 == 0)
        int t0 = e4 & 63;
        float ax0 = sBraw[q * 4 + 0], ay0 = sBraw[q * 4 + 1];
        float ax1 = sBraw[q * 4 + 2], ay1 = sBraw[q * 4 + 3];
        float nx0 = sBn[q * 4 + 0],   ny0 = sBn[q * 4 + 1];
        float nx1 = sBn[q * 4 + 2],   ny1 = sBn[q * 4 + 3];
        float areaA = sAreaA[q];
        float4 cls4 = *(const float4*)(sC + e4);
        float c[4] = {cls4.x, cls4.y, cls4.z, cls4.w};
        float4 res;
        float* rp = (float*)&res;
        #pragma unroll
        for (int k = 0; k < 4; ++k) {
            int t = t0 + k;
            float tx0 = sTraw[t * 4 + 0], ty0 = sTraw[t * 4 + 1];
            float tx1 = sTraw[t * 4 + 2], ty1 = sTraw[t * 4 + 3];

            float l1 = fabsf(nx0 - sTn[t * 4 + 0]) + fabsf(ny0 - sTn[t * 4 + 1])
                     + fabsf(nx1 - sTn[t * 4 + 2]) + fabsf(ny1 - sTn[t * 4 + 3]);

            float iw    = fmaxf(fminf(ax1, tx1) - fmaxf(ax0, tx0), 0.0f);
            float ih    = fmaxf(fminf(ay1, ty1) - fmaxf(ay0, ty0), 0.0f);
            float inter = iw * ih;
            float uni   = areaA + sAreaB[t] - inter;
            float iou   = inter / uni;
            float ew    = fmaxf(fmaxf(ax1, tx1) - fminf(ax0, tx0), 0.0f);
            float eh    = fmaxf(fmaxf(ay1, ty1) - fminf(ay0, ty0), 0.0f);
            float enc   = ew * eh;
            float giou  = iou - (enc - uni) / enc;

            rp[k] = c[k] + 5.0f * l1 - 2.0f * giou;   // L1_W=5, GIOU_W=2 (cost = -giou)
        }
        *(float4*)(sC + e4) = res;                         // ds_store_b128
        *(float4*)(outC + (size_t)b * QT + e4) = res;      // global_store_b128
    }
    __syncthreads();

    // ---- Phase 2: 64 greedy argmin rounds, LDS-resident, b128 scans ----
    const float INF  = __builtin_inff();
    const int   wave = tid >> 5, lane = tid & 31;
    for (int it = 0; it < TT; ++it) {
        // b128 strided scan; strict '<' over increasing flat index keeps first occurrence
        float bvv = INF; int bii = QT;
        for (int e4 = tid * 4; e4 < QT; e4 += NTHREADS * 4) {
            float4 v = *(const float4*)(sC + e4);          // ds_load_b128
            if (v.x < bvv) { bvv = v.x; bii = e4;     }
            if (v.y < bvv) { bvv = v.y; bii = e4 + 1; }
            if (v.z < bvv) { bvv = v.z; bii = e4 + 2; }
            if (v.w < bvv) { bvv = v.w; bii = e4 + 3; }
        }
        // wave32 cross-lane reduce carrying (value, flat idx) lexicographic key
        #pragma unroll
        for (int off = 16; off > 0; off >>= 1) {
            float ov = __shfl_down(bvv, off, 32);
            int   oi = __shfl_down(bii, off, 32);
            if (ov < bvv || (ov == bvv && oi < bii)) { bvv = ov; bii = oi; }
        }
        if (lane == 0) { sRedV[wave] = bvv; sRedI[wave] = bii; }
        __syncthreads();
        if (tid == 0) {
            float fv = sRedV[0]; int fi = sRedI[0];
            #pragma unroll
            for (int w = 1; w < NWAVES; ++w) {
                float wv = sRedV[w]; int wi = sRedI[w];
                if (wv < fv || (wv == fv && wi < fi)) { fv = wv; fi = wi; }
            }
            sRedI[0] = fi;                                   // broadcast
            outRows[b * TT + it] = (float)(fi >> 6);
            outCols[b * TT + it] = (float)(fi & 63);
        }
        __syncthreads();
        int sel = sRedI[0];           // global (min value, min flat idx) == jnp.argmin
        int i = sel >> 6, j = sel & 63;
        // poison selected row & column (== reference's used-mask with +inf)
        if (tid < 16) {
            float4 inf4; inf4.x = INF; inf4.y = INF; inf4.z = INF; inf4.w = INF;
            *(float4*)(sC + i * TT + tid * 4) = inf4;
        }
        for (int q = tid; q < QQ; q += NTHREADS) sC[q * TT + j] = INF;
        __syncthreads();
    }
}

extern "C" void kernel_launch(void* const* d_in, const int* in_sizes, int n_in,
                              void* d_out, int out_size, void* d_ws, size_t ws_size,
                              hipStream_t stream) {
    (void)in_sizes; (void)n_in; (void)out_size; (void)d_ws; (void)ws_size;
    const float* logits = (const float*)d_in[0];
    const float* boxes  = (const float*)d_in[1];
    const int*   ids    = (const int*)d_in[2];
    const float* tbox   = (const float*)d_in[3];
    const float* img    = (const float*)d_in[4];

    float* out     = (float*)d_out;
    float* outC    = out;                              // [B,Q,T]
    float* outRows = out + (size_t)BB * QT;            // [B,T]
    float* outCols = outRows + (size_t)BB * TT;        // [B,T]

    const size_t smem_bytes = (size_t)SMEM_FLOATS * sizeof(float);  // ~148 KB < 320 KB WGP LDS
    hipFuncSetAttribute((const void*)matcher_fused,
                        hipFuncAttributeMaxDynamicSharedMemorySize, (int)smem_bytes);
    matcher_fused<<<dim3(BB), dim3(NTHREADS), smem_bytes, stream>>>(
        logits, boxes, ids, tbox, img, outC, outRows, outCols);
}